// MSABlock_29764123361870
// MI455X (gfx1250) — compile-verified
//
#include <hip/hip_runtime.h>
#include <hip/hip_bf16.h>
#include <math.h>

typedef float v2f __attribute__((ext_vector_type(2)));
typedef float v8f __attribute__((ext_vector_type(8)));

#define S_MSA   32
#define NPROT   512
#define NRES    2048
#define DMODEL  512
#define DMSA    64
#define HMSA    8
#define DHEAD   8
#define HRES    16
#define RANK    16
#define MAXK    512   // max K supported by the LDS-staged GEMM (16*512*4 = 32 KB LDS)

__device__ __forceinline__ float sigm(float x) { return 1.0f / (1.0f + __expf(-x)); }

// Low 32 bits of a generic pointer into LDS = byte offset relative to LDS_BASE
// (ISA 10.2: LDS aperture flat address keeps the LDS offset in addr[31:0]).
__device__ __forceinline__ unsigned lds_off(const void* p) {
    return (unsigned)(uintptr_t)p;
}
// CDNA5 async global->LDS copy (ASYNCcnt-tracked, CDNA5 ISA).
__device__ __forceinline__ void async_load_f32_to_lds(unsigned lds_byte_off, const float* gaddr) {
    asm volatile("global_load_async_to_lds_b32 %0, %1, off"
                 :: "v"(lds_byte_off), "v"(gaddr) : "memory");
}
__device__ __forceinline__ void wait_async_zero() {
    asm volatile("s_wait_asynccnt 0x0" ::: "memory");
}

// ============================================================================
// Panel-tiled fp32 WMMA GEMM: C[M,N] = alpha * A[M,K] @ B[K,N] (+bias[n]) (+=C)
// Block = 128 threads (4 waves). Each block owns a 64x16 output strip: the
// K x 16 B-panel is staged ONCE into LDS (transposed) through the async
// tensor path, then each wave computes one 16x16 tile with
// V_WMMA_F32_16X16X4_F32. K must be a multiple of 4, M a multiple of 64.
// fp32 WMMA layouts (ISA 7.12.2):
//   A 16x4 : lane L -> m = L&15, k-pair base = 2*(L>>4)
//   B 4x16 : lane L -> n = L&15, k-pair base = 2*(L>>4)
//   D 16x16: vgpr r, lanes 0-15 -> M=r, lanes 16-31 -> M=8+r; N = L&15
// ============================================================================
__global__ void gemm_wmma_f32(const float* __restrict__ A, const float* __restrict__ B,
                              const float* __restrict__ bias, float* __restrict__ C,
                              int K, int lda, int ldb, int ldc,
                              float alpha, int accumulate)
{
    __shared__ float bsT[16 * MAXK];           // B panel, transposed: bsT[col*K + row]
    const int tn  = blockIdx.x * 16;
    const int tm0 = blockIdx.y * 64;
    const int tid = threadIdx.x;               // 0..127

    // ---- stage B panel (coalesced: 16 consecutive lanes hit 16 consecutive floats)
    for (int t = tid; t < 16 * K; t += 128) {  // 16*K is a multiple of 128 for all uses
        const int row = t >> 4;
        const int col = t & 15;
        async_load_f32_to_lds(lds_off(&bsT[col * K + row]),
                              B + (size_t)row * ldb + tn + col);
    }
    wait_async_zero();
    __syncthreads();

    const int wave = tid >> 5;
    const int lane = tid & 31;
    const int l15  = lane & 15;
    const int kh   = (lane >> 4) * 2;
    const int tm   = tm0 + wave * 16;

    v8f acc = {};
    const float* arow = A + (size_t)(tm + l15) * lda;
    const float* bcol = &bsT[l15 * K];
    for (int k = 0; k < K; k += 4) {
        v2f a = *(const v2f*)(arow + k + kh);   // global_load_b64
        v2f b = *(const v2f*)(bcol + k + kh);   // ds_load_b64
        acc = __builtin_amdgcn_wmma_f32_16x16x4_f32(false, a, false, b,
                                                    (short)0, acc, false, false);
    }
    const int n    = tn + l15;
    const int half = lane >> 4;
    const float bv = bias ? bias[n] : 0.0f;
    for (int r = 0; r < 8; ++r) {
        const int m = tm + half * 8 + r;
        size_t o    = (size_t)m * ldc + n;
        float val   = alpha * acc[r] + bv;
        if (accumulate) val += C[o];
        C[o] = val;
    }
}

// ============================================================================
// Fused SwiGLU GEMM: C = silu(A@B1) * (A@B3), K = DMSA (64), panel-tiled like
// above with both weight panels staged asynchronously into LDS.
// ============================================================================
__global__ void gemm_swiglu_wmma(const float* __restrict__ A, const float* __restrict__ B1,
                                 const float* __restrict__ B3, float* __restrict__ C,
                                 int K, int lda, int ldb, int ldc)
{
    __shared__ float b1T[16 * DMSA];
    __shared__ float b3T[16 * DMSA];
    const int tn  = blockIdx.x * 16;
    const int tm0 = blockIdx.y * 64;
    const int tid = threadIdx.x;

    for (int t = tid; t < 16 * K; t += 128) {
        const int row = t >> 4;
        const int col = t & 15;
        async_load_f32_to_lds(lds_off(&b1T[col * K + row]), B1 + (size_t)row * ldb + tn + col);
        async_load_f32_to_lds(lds_off(&b3T[col * K + row]), B3 + (size_t)row * ldb + tn + col);
    }
    wait_async_zero();
    __syncthreads();

    const int wave = tid >> 5;
    const int lane = tid & 31;
    const int l15  = lane & 15;
    const int kh   = (lane >> 4) * 2;
    const int tm   = tm0 + wave * 16;

    v8f acc1 = {}, acc3 = {};
    const float* arow = A + (size_t)(tm + l15) * lda;
    const float* b1c  = &b1T[l15 * K];
    const float* b3c  = &b3T[l15 * K];
    for (int k = 0; k < K; k += 4) {
        v2f a  = *(const v2f*)(arow + k + kh);
        v2f b1 = *(const v2f*)(b1c + k + kh);
        v2f b3 = *(const v2f*)(b3c + k + kh);
        acc1 = __builtin_amdgcn_wmma_f32_16x16x4_f32(false, a, false, b1,
                                                     (short)0, acc1, false, false);
        acc3 = __builtin_amdgcn_wmma_f32_16x16x4_f32(false, a, false, b3,
                                                     (short)0, acc3, false, false);
    }
    const int n    = tn + l15;
    const int half = lane >> 4;
    for (int r = 0; r < 8; ++r) {
        const int m = tm + half * 8 + r;
        float t1 = acc1[r];
        C[(size_t)m * ldc + n] = (t1 * sigm(t1)) * acc3[r];
    }
}

// ============================================================================
// Row attention (per s, h, 16-row i tile). Computes transposed score tiles
// S^T[j][i] = K_tile(16x8) @ Q_tile^T(8x16) with WMMA, online softmax over j,
// VALU P@V accumulation (D_HEAD = 8). One wave per block.
// ============================================================================
__global__ void msa_row_attention(const float* __restrict__ qb, const float* __restrict__ kb,
                                  const float* __restrict__ vb, const float* __restrict__ pos_bias,
                                  const float* __restrict__ padmask, float* __restrict__ att_out)
{
    const int i0   = blockIdx.x * 16;
    const int h    = blockIdx.y;
    const int s    = blockIdx.z;
    const int lane = threadIdx.x & 31;
    const int l15  = lane & 15;
    const int half = lane >> 4;
    const int kh   = half * 2;
    const float qk_scale = 0.35355339059327373f; // 1/sqrt(8)

    // Q fragments (B role): column i = i0 + l15, constant across the j loop.
    const size_t qbase = ((size_t)s * NPROT + i0 + l15) * DMSA + h * DHEAD;
    v2f bq0 = *(const v2f*)(qb + qbase + kh);
    v2f bq1 = *(const v2f*)(qb + qbase + 4 + kh);

    float mrun = -1e30f, srun = 0.0f;
    float acc_o[DHEAD];
    for (int d = 0; d < DHEAD; ++d) acc_o[d] = 0.0f;

    const size_t pbrow = ((size_t)h * NPROT + i0 + l15) * NPROT;

    for (int j0 = 0; j0 < NPROT; j0 += 16) {
        const size_t kbase = ((size_t)s * NPROT + j0 + l15) * DMSA + h * DHEAD;
        v2f a0 = *(const v2f*)(kb + kbase + kh);
        v2f a1 = *(const v2f*)(kb + kbase + 4 + kh);

        v8f st = {};
        st = __builtin_amdgcn_wmma_f32_16x16x4_f32(false, a0, false, bq0, (short)0, st, false, false);
        st = __builtin_amdgcn_wmma_f32_16x16x4_f32(false, a1, false, bq1, (short)0, st, false, false);

        float stv[8];
        float tmax = -1e30f;
        for (int r = 0; r < 8; ++r) {
            const int jj = j0 + half * 8 + r;
            float v = st[r] * qk_scale + pos_bias[pbrow + jj]
                      + (1.0f - padmask[jj]) * (-1e9f);
            stv[r] = v;
            tmax = fmaxf(tmax, v);
        }
        tmax = fmaxf(tmax, __shfl_xor(tmax, 16, 32));
        const float mnew = fmaxf(mrun, tmax);
        const float rs   = __expf(mrun - mnew);

        float p[8];
        float tsum = 0.0f;
        for (int r = 0; r < 8; ++r) { p[r] = __expf(stv[r] - mnew); tsum += p[r]; }
        tsum += __shfl_xor(tsum, 16, 32);
        srun = srun * rs + tsum;
        for (int d = 0; d < DHEAD; ++d) acc_o[d] *= rs;
        mrun = mnew;

        for (int r = 0; r < 8; ++r) {
            const size_t vbase = ((size_t)s * NPROT + j0 + half * 8 + r) * DMSA + h * DHEAD;
            const float pr = p[r];
            for (int d = 0; d < DHEAD; ++d) acc_o[d] += pr * vb[vbase + d];
        }
    }

    const float inv = 1.0f / srun;
    for (int d = 0; d < DHEAD; ++d) {
        float v = acc_o[d] + __shfl_xor(acc_o[d], 16, 32);
        if (lane < 16)
            att_out[((size_t)s * NPROT + i0 + lane) * DMSA + h * DHEAD + d] = v * inv;
    }
}

// ============================================================================
// Coevolution: per 16-row i tile, loop j tiles; for each rank r compute the
// 16x16 tile of c_r = (1/S) U_r @ V_r^T via WMMA (K = S = 32), accumulate
// w = sigmoid(sum_r c*w_cw + b_cw)*padmask_j  and c_bar row sums.
// ============================================================================
__global__ void coevol_kernel(const float* __restrict__ u, const float* __restrict__ v,
                              const float* __restrict__ w_cw, const float* __restrict__ b_cw_p,
                              const float* __restrict__ padmask, const int* __restrict__ idx,
                              float* __restrict__ w_mat, float* __restrict__ cbar_g)
{
    __shared__ float cbar_s[16][RANK];
    const int i0   = blockIdx.x * 16;
    const int lane = threadIdx.x & 31;
    const int l15  = lane & 15;
    const int half = lane >> 4;
    const int kh   = half * 2;
    const float invS = 1.0f / (float)S_MSA;
    const float bcw  = b_cw_p[0];

    for (int t = lane; t < 16 * RANK; t += 32) (&cbar_s[0][0])[t] = 0.0f;
    __syncthreads();

    for (int j0 = 0; j0 < NPROT; j0 += 16) {
        float wsum[8];
        for (int q = 0; q < 8; ++q) wsum[q] = bcw;

        for (int r = 0; r < RANK; ++r) {
            v8f acc = {};
            for (int k = 0; k < S_MSA; k += 4) {
                v2f a, b;
                const int s0 = k + kh;
                a.x = u[((size_t)s0 * NPROT + i0 + l15) * RANK + r];
                a.y = u[((size_t)(s0 + 1) * NPROT + i0 + l15) * RANK + r];
                b.x = v[((size_t)s0 * NPROT + j0 + l15) * RANK + r];
                b.y = v[((size_t)(s0 + 1) * NPROT + j0 + l15) * RANK + r];
                acc = __builtin_amdgcn_wmma_f32_16x16x4_f32(false, a, false, b,
                                                            (short)0, acc, false, false);
            }
            const float wr = w_cw[r];
            float rsum[8];
            for (int q = 0; q < 8; ++q) {
                const float cv = acc[q] * invS;
                wsum[q] += wr * cv;
                float rr = cv;
                rr += __shfl_xor(rr, 1, 32);
                rr += __shfl_xor(rr, 2, 32);
                rr += __shfl_xor(rr, 4, 32);
                rr += __shfl_xor(rr, 8, 32);
                rsum[q] = rr;   // full 16-column sum, valid in every lane of the half
            }
            if (l15 == 0) {
                for (int q = 0; q < 8; ++q)
                    cbar_s[half * 8 + q][r] += rsum[q];
            }
        }
        const float pm = padmask[j0 + l15];
        for (int q = 0; q < 8; ++q) {
            const float wv = sigm(wsum[q]) * pm;
            w_mat[(size_t)(i0 + half * 8 + q) * NPROT + j0 + l15] = wv;
        }
    }
    __syncthreads();
    // c_bar_prot[i][r] = (sum_j c) * padmask[i]; scatter /NPROT into (NRES,RANK)
    for (int t = lane; t < 16 * RANK; t += 32) {
        const int row = t / RANK, r = t % RANK;
        const int gi  = i0 + row;
        cbar_g[(size_t)idx[gi] * RANK + r] =
            cbar_s[row][r] * padmask[gi] * (1.0f / (float)NPROT);
    }
}

// ============================================================================
// LayerNorm over last dim d
// ============================================================================
__global__ void layernorm_rows(const float* __restrict__ x, const float* __restrict__ g,
                               const float* __restrict__ b, float* __restrict__ y, int d)
{
    __shared__ float red[256];
    const int row = blockIdx.x;
    const int tid = threadIdx.x;
    const int T   = blockDim.x;
    const float* xr = x + (size_t)row * d;
    float* yr = y + (size_t)row * d;

    float s = 0.0f;
    for (int c = tid; c < d; c += T) s += xr[c];
    red[tid] = s; __syncthreads();
    for (int off = T >> 1; off > 0; off >>= 1) {
        if (tid < off) red[tid] += red[tid + off];
        __syncthreads();
    }
    const float mean = red[0] / (float)d;
    __syncthreads();

    float vv = 0.0f;
    for (int c = tid; c < d; c += T) { float t = xr[c] - mean; vv += t * t; }
    red[tid] = vv; __syncthreads();
    for (int off = T >> 1; off > 0; off >>= 1) {
        if (tid < off) red[tid] += red[tid + off];
        __syncthreads();
    }
    const float rstd = rsqrtf(red[0] / (float)d + 1e-5f);
    for (int c = tid; c < d; c += T)
        yr[c] = (xr[c] - mean) * rstd * g[c] + b[c];
}

// ============================================================================
// Column aggregation: alpha = softmax_s(m_c . w_colw); col_agg = sum_s alpha*m_c
// ============================================================================
__global__ void col_aggregate(const float* __restrict__ mc, const float* __restrict__ w_colw,
                              float* __restrict__ col_agg)
{
    const int i = blockIdx.x;
    const int s = threadIdx.x & 31;
    const float* row = mc + ((size_t)s * NPROT + i) * DMSA;
    float dot = 0.0f;
    for (int d = 0; d < DMSA; ++d) dot += row[d] * w_colw[d];
    float m = dot;
    for (int k = 16; k > 0; k >>= 1) m = fmaxf(m, __shfl_xor(m, k, 32));
    const float e = __expf(dot - m);
    float ss = e;
    for (int k = 16; k > 0; k >>= 1) ss += __shfl_xor(ss, k, 32);
    const float alpha = e / ss;
    for (int d = 0; d < DMSA; ++d) {
        float v = alpha * row[d];
        for (int k = 16; k > 0; k >>= 1) v += __shfl_xor(v, k, 32);
        if (s == 0) col_agg[(size_t)i * DMSA + d] = v;
    }
}

// ============================================================================
// Marginal softmax over N for one head
// ============================================================================
__global__ void marginal_softmax(const float* __restrict__ logit, const float* __restrict__ cbias,
                                 const float* __restrict__ alpha_coevol,
                                 const unsigned char* __restrict__ protmask,
                                 float* __restrict__ out)
{
    __shared__ float red[256];
    const int h   = blockIdx.x;
    const int tid = threadIdx.x;
    const float ah = alpha_coevol[h];
    float x[NRES / 256];
    float lm = -1e30f;
    for (int kk = 0; kk < NRES / 256; ++kk) {
        const int n = tid + kk * 256;
        const float cb = cbias[(size_t)n * HRES + h] * (protmask[n] ? 1.0f : 0.0f);
        const float v  = logit[(size_t)n * HRES + h] + ah * cb;
        x[kk] = v; lm = fmaxf(lm, v);
    }
    red[tid] = lm; __syncthreads();
    for (int off = 128; off > 0; off >>= 1) {
        if (tid < off) red[tid] = fmaxf(red[tid], red[tid + off]);
        __syncthreads();
    }
    const float mx = red[0]; __syncthreads();
    float ls = 0.0f;
    for (int kk = 0; kk < NRES / 256; ++kk) { x[kk] = __expf(x[kk] - mx); ls += x[kk]; }
    red[tid] = ls; __syncthreads();
    for (int off = 128; off > 0; off >>= 1) {
        if (tid < off) red[tid] += red[tid + off];
        __syncthreads();
    }
    const float inv = 1.0f / red[0];
    for (int kk = 0; kk < NRES / 256; ++kk)
        out[(size_t)h * NRES + tid + kk * 256] = x[kk] * inv;
}

// ---------------------------------------------------------------------------
// Small helper kernels
// ---------------------------------------------------------------------------
__global__ void build_idx(const unsigned char* __restrict__ mask, int* __restrict__ idx)
{
    if (threadIdx.x == 0 && blockIdx.x == 0) {
        int cnt = 0;
        for (int n = 0; n < NRES && cnt < NPROT; ++n)
            if (mask[n]) idx[cnt++] = n;
        for (int n = 0; n < NRES && cnt < NPROT; ++n)   // non-protein tail (stable argsort)
            if (!mask[n]) idx[cnt++] = n;
    }
}

__global__ void gather_rows(const float* __restrict__ src, const int* __restrict__ idx,
                            float* __restrict__ dst, int d)
{
    const int i = blockIdx.x;
    const int row = idx[i];
    for (int c = threadIdx.x; c < d; c += blockDim.x)
        dst[(size_t)i * d + c] = src[(size_t)row * d + c];
}

__global__ void scatter_add_rows(float* __restrict__ dst, const int* __restrict__ idx,
                                 const float* __restrict__ src, int d)
{
    const int i = blockIdx.x;
    const int row = idx[i];
    for (int c = threadIdx.x; c < d; c += blockDim.x)
        dst[(size_t)row * d + c] += src[(size_t)i * d + c];
}

__global__ void add_broadcast_sprot(float* __restrict__ m, const float* __restrict__ hprot, int total)
{
    const int t = blockIdx.x * blockDim.x + threadIdx.x;
    if (t < total) m[t] += hprot[t & (NPROT * DMSA - 1)];
}

__global__ void gate_add(float* __restrict__ m, const float* __restrict__ g,
                         const float* __restrict__ o, int total)
{
    const int t = blockIdx.x * blockDim.x + threadIdx.x;
    if (t < total) m[t] += sigm(g[t]) * o[t];
}

__global__ void bias_rows_add(float* __restrict__ h, const float* __restrict__ bias, int total)
{
    const int t = blockIdx.x * blockDim.x + threadIdx.x;
    if (t < total) h[t] += bias[t & (DMODEL - 1)];
}

__global__ void zero_buf(float* __restrict__ p, int n)
{
    const int t = blockIdx.x * blockDim.x + threadIdx.x;
    if (t < n) p[t] = 0.0f;
}

// ============================================================================
// Host-side orchestration
// ============================================================================
extern "C" void kernel_launch(void* const* d_in, const int* in_sizes, int n_in,
                              void* d_out, int out_size, void* d_ws, size_t ws_size,
                              hipStream_t stream)
{
    const float* in_m          = (const float*)d_in[0];
    const float* in_h          = (const float*)d_in[1];
    const unsigned char* pmask = (const unsigned char*)d_in[4]; // numpy bool_, 1 byte
    const float* pos_bias      = (const float*)d_in[5];
    const float* alpha_coev    = (const float*)d_in[6];
    const float* padmask       = (const float*)d_in[7];
    #define PARAM(i) ((const float*)d_in[i])
    const float *w_s2m = PARAM(8),  *b_s2m = PARAM(9);
    const float *g_row = PARAM(10), *b_row = PARAM(11);
    const float *wq = PARAM(12), *wk = PARAM(13), *wv = PARAM(14), *wg = PARAM(15), *wo = PARAM(16);
    const float *g_col = PARAM(17), *b_col = PARAM(18);
    const float *w_colw = PARAM(19), *w_c2s = PARAM(20);
    const float *g_coev = PARAM(21), *b_coev = PARAM(22);
    const float *w_u = PARAM(23), *w_v = PARAM(24);
    const float *g_cval = PARAM(25), *b_cval = PARAM(26);
    const float *w_cval = PARAM(27), *b_cval2 = PARAM(28);
    const float *w_cw = PARAM(29), *b_cw = PARAM(30);
    const float *w_cout = PARAM(31), *b_cout = PARAM(32);
    const float *w_mu = PARAM(33), *b_mu = PARAM(34);
    const float *w_nu = PARAM(35), *b_nu = PARAM(36);
    const float *w_c2m = PARAM(37), *b_c2m = PARAM(38);
    const float *g_ff = PARAM(39), *b_ff = PARAM(40);
    const float *w1 = PARAM(41), *w3 = PARAM(42), *w2 = PARAM(43);

    float* out = (float*)d_out;
    const size_t M_ELEMS = (size_t)S_MSA * NPROT * DMSA;   // 1,048,576
    const size_t H_ELEMS = (size_t)NRES * DMODEL;          // 1,048,576

    float* ws = (float*)d_ws;
    size_t off = 0;
    auto alloc = [&](size_t n) { float* p = ws + off; off += n; return p; };
    int*   idx    = (int*)alloc(512);
    float* hres   = alloc(H_ELEMS);
    float* mbuf   = alloc(M_ELEMS);
    float* lnb    = alloc(M_ELEMS);
    float* qb     = alloc(M_ELEMS);
    float* kb     = alloc(M_ELEMS);
    float* vb     = alloc(M_ELEMS);
    float* gb     = alloc(M_ELEMS);
    float* att    = alloc(M_ELEMS);
    float* tmp1   = alloc((size_t)S_MSA * NPROT * 4 * DMSA); // SwiGLU hidden / wo result
    float* hcn    = alloc(H_ELEMS);
    float* hcoev  = alloc(H_ELEMS);
    float* hsel   = alloc((size_t)NPROT * DMODEL);
    float* hj     = alloc((size_t)NPROT * DMODEL);
    float* ub     = alloc((size_t)S_MSA * NPROT * RANK);
    float* vb2    = alloc((size_t)S_MSA * NPROT * RANK);
    float* wmat   = alloc((size_t)NPROT * NPROT);
    float* cbar   = alloc((size_t)NRES * RANK);
    float* hprot  = alloc((size_t)NPROT * DMSA);
    float* colagg = alloc((size_t)NPROT * DMSA);
    float* delta  = alloc((size_t)NPROT * DMODEL);
    float* hjagg  = alloc((size_t)NPROT * DMODEL);
    float* mul    = alloc((size_t)NRES * HRES);
    float* nul    = alloc((size_t)NRES * HRES);
    float* cbias  = alloc((size_t)NRES * HRES);
    (void)ws_size; (void)n_in; (void)in_sizes; (void)out_size;

    // panel GEMM: M % 64 == 0, K <= MAXK, K % 4 == 0
    auto gemm = [&](const float* A, const float* B, const float* bias, float* C,
                    int M, int N, int K, int lda, int ldb, int ldc, float alpha, int acc) {
        gemm_wmma_f32<<<dim3(N / 16, M / 64), 128, 0, stream>>>(A, B, bias, C, K, lda, ldb, ldc, alpha, acc);
    };

    hipMemcpyAsync(hres, in_h, H_ELEMS * sizeof(float), hipMemcpyDeviceToDevice, stream);
    hipMemcpyAsync(mbuf, in_m, M_ELEMS * sizeof(float), hipMemcpyDeviceToDevice, stream);

    // ---- idx = first NPROT protein positions (stable argsort semantics)
    build_idx<<<1, 32, 0, stream>>>(pmask, idx);

    // ---- m += gather(h_res @ w_s2m + b_s2m, idx)
    gather_rows<<<NPROT, 256, 0, stream>>>(hres, idx, hsel, DMODEL);
    gemm(hsel, w_s2m, b_s2m, hprot, NPROT, DMSA, DMODEL, DMODEL, DMSA, DMSA, 1.0f, 0);
    add_broadcast_sprot<<<(int)(M_ELEMS / 256), 256, 0, stream>>>(mbuf, hprot, (int)M_ELEMS);

    // ---- masked row attention with positional bias
    layernorm_rows<<<S_MSA * NPROT, 64, 0, stream>>>(mbuf, g_row, b_row, lnb, DMSA);
    const int MR = S_MSA * NPROT; // 16384 rows of d_msa
    gemm(lnb, wq, nullptr, qb, MR, DMSA, DMSA, DMSA, DMSA, DMSA, 1.0f, 0);
    gemm(lnb, wk, nullptr, kb, MR, DMSA, DMSA, DMSA, DMSA, DMSA, 1.0f, 0);
    gemm(lnb, wv, nullptr, vb, MR, DMSA, DMSA, DMSA, DMSA, DMSA, 1.0f, 0);
    gemm(lnb, wg, nullptr, gb, MR, DMSA, DMSA, DMSA, DMSA, DMSA, 1.0f, 0);
    msa_row_attention<<<dim3(NPROT / 16, HMSA, S_MSA), 32, 0, stream>>>(qb, kb, vb, pos_bias, padmask, att);
    gemm(att, wo, nullptr, tmp1, MR, DMSA, DMSA, DMSA, DMSA, DMSA, 1.0f, 0);
    gate_add<<<(int)(M_ELEMS / 256), 256, 0, stream>>>(mbuf, gb, tmp1, (int)M_ELEMS);

    // ---- column aggregation -> scatter-add into h_res
    layernorm_rows<<<S_MSA * NPROT, 64, 0, stream>>>(mbuf, g_col, b_col, lnb, DMSA);
    col_aggregate<<<NPROT, 32, 0, stream>>>(lnb, w_colw, colagg);
    gemm(colagg, w_c2s, nullptr, delta, NPROT, DMODEL, DMSA, DMSA, DMODEL, DMODEL, 1.0f, 0);
    scatter_add_rows<<<NPROT, 256, 0, stream>>>(hres, idx, delta, DMODEL);

    // ---- low-rank coevolution
    layernorm_rows<<<S_MSA * NPROT, 64, 0, stream>>>(mbuf, g_coev, b_coev, lnb, DMSA);
    gemm(lnb, w_u, nullptr, ub,  MR, RANK, DMSA, DMSA, RANK, RANK, 1.0f, 0);
    gemm(lnb, w_v, nullptr, vb2, MR, RANK, DMSA, DMSA, RANK, RANK, 1.0f, 0);
    layernorm_rows<<<NRES, 256, 0, stream>>>(hres, g_cval, b_cval, hcn, DMODEL);
    gemm(hcn, w_cval, b_cval2, hcoev, NRES, DMODEL, DMODEL, DMODEL, DMODEL, DMODEL, 1.0f, 0);
    gather_rows<<<NPROT, 256, 0, stream>>>(hcoev, idx, hj, DMODEL);
    zero_buf<<<(NRES * RANK) / 256, 256, 0, stream>>>(cbar, NRES * RANK);
    coevol_kernel<<<NPROT / 16, 32, 0, stream>>>(ub, vb2, w_cw, b_cw, padmask, idx, wmat, cbar);
    gemm(wmat, hj, nullptr, hjagg, NPROT, DMODEL, NPROT, NPROT, DMODEL, DMODEL, 1.0f, 0);
    gemm(hjagg, w_cout, nullptr, delta, NPROT, DMODEL, DMODEL, DMODEL, DMODEL, DMODEL,
         1.0f / (float)NPROT, 0);
    scatter_add_rows<<<NPROT, 256, 0, stream>>>(hres, idx, delta, DMODEL);
    bias_rows_add<<<(int)(H_ELEMS / 256), 256, 0, stream>>>(hres, b_cout, (int)H_ELEMS);

    // ---- marginal updates (outputs 3 and 4)
    gemm(hres, w_mu, b_mu, mul, NRES, HRES, DMODEL, DMODEL, HRES, HRES, 1.0f, 0);
    gemm(hres, w_nu, b_nu, nul, NRES, HRES, DMODEL, DMODEL, HRES, HRES, 1.0f, 0);
    gemm(cbar, w_c2m, b_c2m, cbias, NRES, HRES, RANK, RANK, HRES, HRES, 1.0f, 0);
    marginal_softmax<<<HRES, 256, 0, stream>>>(mul, cbias, alpha_coev, pmask, out + 2 * M_ELEMS);
    marginal_softmax<<<HRES, 256, 0, stream>>>(nul, cbias, alpha_coev, pmask,
                                               out + 2 * M_ELEMS + (size_t)HRES * NRES);

    // ---- SwiGLU transition
    layernorm_rows<<<S_MSA * NPROT, 64, 0, stream>>>(mbuf, g_ff, b_ff, lnb, DMSA);
    gemm_swiglu_wmma<<<dim3((4 * DMSA) / 16, MR / 64), 128, 0, stream>>>(lnb, w1, w3, tmp1,
                                                                         DMSA, DMSA, 4 * DMSA, 4 * DMSA);
    gemm(tmp1, w2, nullptr, mbuf, MR, DMSA, 4 * DMSA, 4 * DMSA, DMSA, DMSA, 1.0f, 1);

    // ---- outputs: m ‖ h_res ‖ mu ‖ nu
    hipMemcpyAsync(out, mbuf, M_ELEMS * sizeof(float), hipMemcpyDeviceToDevice, stream);
    hipMemcpyAsync(out + M_ELEMS, hres, H_ELEMS * sizeof(float), hipMemcpyDeviceToDevice, stream);
}